// AGG_7438883357375
// MI455X (gfx1250) — compile-verified
//
#include <hip/hip_runtime.h>
#include <math.h>

#define B_   32
#define N_   16384
#define TED  64
#define FPD  64
#define HID  128
#define OUT_ 64

typedef __attribute__((ext_vector_type(2))) float v2f;
typedef __attribute__((ext_vector_type(8))) float v8f;

// workspace layout (float offsets)
#define WS_U     0          // [128]
#define WS_C     128        // [128]
#define WS_BASE  256        // [32][128]  h_query + c
#define WS_STATS 4352       // [32][2]    softmax max,sum
#define WS_TBAR  4416       // [32][65]   att-weighted tau sums + s_kv

// ---------------------------------------------------------------- phase 0
__global__ void k0_prep(const float* __restrict__ qt,
                        const float* __restrict__ t2v_w, const float* __restrict__ t2v_b,
                        const float* __restrict__ w_proj, const float* __restrict__ b_proj,
                        const float* __restrict__ w_query, const float* __restrict__ w_key,
                        float* __restrict__ ws) {
  __shared__ float qv[B_ * TED];
  const int tid = threadIdx.x;                       // 128 threads
  for (int idx = tid; idx < B_ * TED; idx += 128) {
    int b = idx / TED, k = idx % TED;
    float v = fmaf(qt[b], t2v_w[k], t2v_b[k]);
    qv[idx] = (k == 0) ? v : __sinf(v);
  }
  for (int idx = tid; idx < B_ * 65; idx += 128) ws[WS_TBAR + idx] = 0.f;
  __syncthreads();
  const int h = tid;                                 // 0..127
  float u = 0.f, c = 0.f;
  for (int f = 0; f < FPD; ++f) {
    float wk = w_key[f * HID + h];
    u = fmaf(w_proj[f], wk, u);
    c = fmaf(b_proj[f], wk, c);
  }
  ws[WS_U + h] = u;
  ws[WS_C + h] = c;
  for (int b = 0; b < B_; ++b) {
    float hq = 0.f;
    for (int k = 0; k < TED; ++k) hq = fmaf(qv[b * TED + k], w_query[k * HID + h], hq);
    ws[WS_BASE + b * HID + h] = hq + c;
  }
}

// ---------------------------------------------------------------- phase 1: WMMA GEMM + logits
__global__ void __launch_bounds__(256)
k1_logits(const float* __restrict__ key_value, const float* __restrict__ key_time,
          const float* __restrict__ t2v_w, const float* __restrict__ t2v_b,
          const float* __restrict__ w_key, const float* __restrict__ w_attn,
          const float* __restrict__ ws, float* __restrict__ e_out) {
  __shared__ float W2[TED * HID];                    // rows 64..127 of w_key, 32 KB
  __shared__ float tw[TED], tb[TED];
  const int tid = threadIdx.x;
  for (int idx = tid; idx < TED * HID; idx += 256) W2[idx] = w_key[FPD * HID + idx];
  if (tid < TED) { tw[tid] = t2v_w[tid]; tb[tid] = t2v_b[tid]; }
  __syncthreads();

  const int wave = tid >> 5, lane = tid & 31;
  const int half = lane >> 4, col = lane & 15;
  const int b  = blockIdx.y;
  const int n0 = (blockIdx.x * 8 + wave) * 16;       // 16-row n tile per wave
  const float* ktp = key_time  + b * N_ + n0;
  const float* kvp = key_value + b * N_ + n0;

  // A fragments: 16x4 fp32 tiles of tau. Lane layout per ISA 7.12.2:
  // lanes 0-15: M=lane, K={4j,4j+1}; lanes 16-31: M=lane-16, K={4j+2,4j+3}
  const float kt = ktp[col];
  const int kofs = half * 2;
  v2f a[16];
#pragma unroll
  for (int j = 0; j < 16; ++j) {
    int k0 = 4 * j + kofs;
    float v0 = fmaf(kt, tw[k0],     tb[k0]);
    float v1 = fmaf(kt, tw[k0 + 1], tb[k0 + 1]);
    float s0 = __sinf(v0);
    a[j].x = (k0 == 0) ? v0 : s0;                    // time2vec: k==0 linear
    a[j].y = __sinf(v1);
  }
  float kvv[8];
#pragma unroll
  for (int i = 0; i < 8; ++i) kvv[i] = kvp[8 * half + i];

  float ep[8] = {0.f, 0.f, 0.f, 0.f, 0.f, 0.f, 0.f, 0.f};
  for (int t = 0; t < 8; ++t) {                      // 8 h-tiles of 16
    // preload all 16 B fragments for this tile -> single dscnt wait
    v2f bf[16];
#pragma unroll
    for (int j = 0; j < 16; ++j) {
      int kk = 4 * j + kofs;                         // B: lanes 0-15 K={kk,kk+1}, lanes 16-31 K+2
      bf[j].x = W2[kk * HID + 16 * t + col];
      bf[j].y = W2[(kk + 1) * HID + 16 * t + col];
    }
    // two independent accumulator chains for matrix-pipe ILP
    v8f acc0 = {0.f, 0.f, 0.f, 0.f, 0.f, 0.f, 0.f, 0.f};
    v8f acc1 = {0.f, 0.f, 0.f, 0.f, 0.f, 0.f, 0.f, 0.f};
#pragma unroll
    for (int j = 0; j < 8; ++j) {
      acc0 = __builtin_amdgcn_wmma_f32_16x16x4_f32(false, a[2 * j],     false, bf[2 * j],
                                                   (short)0, acc0, false, false);
      acc1 = __builtin_amdgcn_wmma_f32_16x16x4_f32(false, a[2 * j + 1], false, bf[2 * j + 1],
                                                   (short)0, acc1, false, false);
    }
    // C layout: vgpr i, lanes0-15 -> (M=i, N=lane); lanes16-31 -> (M=8+i, N=lane-16)
    const int h = 16 * t + col;
    const float uh = ws[WS_U + h];
    const float bh = ws[WS_BASE + b * HID + h];
    const float ah = w_attn[h];
#pragma unroll
    for (int i = 0; i < 8; ++i) {
      float pre = (acc0[i] + acc1[i]) + fmaf(kvv[i], uh, bh);
      float act = (pre >= 0.f) ? pre : 0.2f * pre;   // leaky relu
      ep[i] = fmaf(act, ah, ep[i]);
    }
  }
  // reduce over the 16 columns held within each half-wave
#pragma unroll
  for (int i = 0; i < 8; ++i) {
    float v = ep[i];
    v += __shfl_xor(v, 1, 32);
    v += __shfl_xor(v, 2, 32);
    v += __shfl_xor(v, 4, 32);
    v += __shfl_xor(v, 8, 32);
    ep[i] = v;
  }
  if (col == 0) {
#pragma unroll
    for (int i = 0; i < 8; ++i) e_out[b * N_ + n0 + 8 * half + i] = ep[i];
  }
}

// ---------------------------------------------------------------- phase 2: softmax stats
__global__ void k2_stats(const float* __restrict__ e, float* __restrict__ ws) {
  __shared__ float red[256];
  const int b = blockIdx.x, tid = threadIdx.x;
  const float* ep = e + b * N_;
  float m = -INFINITY;
  for (int n = tid; n < N_; n += 256) m = fmaxf(m, ep[n]);
  red[tid] = m; __syncthreads();
  for (int s = 128; s > 0; s >>= 1) { if (tid < s) red[tid] = fmaxf(red[tid], red[tid + s]); __syncthreads(); }
  m = red[0]; __syncthreads();
  float s = 0.f;
  for (int n = tid; n < N_; n += 256) s += __expf(ep[n] - m);
  red[tid] = s; __syncthreads();
  for (int st = 128; st > 0; st >>= 1) { if (tid < st) red[tid] += red[tid + st]; __syncthreads(); }
  if (tid == 0) { ws[WS_STATS + 2 * b] = m; ws[WS_STATS + 2 * b + 1] = red[0]; }
}

// ---------------------------------------------------------------- phase 3: att + weighted tau sums
__global__ void __launch_bounds__(256)
k3_att(const float* __restrict__ key_value, const float* __restrict__ key_time,
       const float* __restrict__ t2v_w, const float* __restrict__ t2v_b,
       float* __restrict__ ws, float* __restrict__ att) {
  __shared__ float sacc[65];
  __shared__ float tw[TED], tb[TED];
  const int tid = threadIdx.x;
  const int b = blockIdx.y, chunk = blockIdx.x;      // 8 chunks of 2048 per b
  if (tid < 65) sacc[tid] = 0.f;
  if (tid < TED) { tw[tid] = t2v_w[tid]; tb[tid] = t2v_b[tid]; }
  __syncthreads();
  const float m   = ws[WS_STATS + 2 * b];
  const float inv = 1.0f / ws[WS_STATS + 2 * b + 1];
  float acc[TED];
#pragma unroll
  for (int k = 0; k < TED; ++k) acc[k] = 0.f;
  float ackv = 0.f;
  const int nbase = b * N_ + chunk * 2048;
  for (int i = tid; i < 2048; i += 256) {
    const int n = nbase + i;
    float aw = __expf(att[n] - m) * inv;             // att[] currently holds e
    att[n] = aw;                                     // overwrite with final attention
    ackv = fmaf(aw, key_value[n], ackv);
    const float kt = key_time[n];
#pragma unroll
    for (int k = 0; k < TED; ++k) {
      float v  = fmaf(kt, tw[k], tb[k]);
      float tv = __sinf(v);
      if (k == 0) tv = v;
      acc[k] = fmaf(aw, tv, acc[k]);
    }
  }
#pragma unroll
  for (int k = 0; k < TED; ++k) atomicAdd(&sacc[k], acc[k]);
  atomicAdd(&sacc[64], ackv);
  __syncthreads();
  if (tid < 65) atomicAdd(&ws[WS_TBAR + b * 65 + tid], sacc[tid]);
}

// ---------------------------------------------------------------- phase 4: scored -> ELU -> y
__global__ void k4_out(const float* __restrict__ w_key, const float* __restrict__ w_pred,
                       const float* __restrict__ b_pred, const float* __restrict__ ws,
                       float* __restrict__ y) {
  __shared__ float hp[B_ * HID];
  const int tid = threadIdx.x;
  for (int idx = tid; idx < B_ * HID; idx += 256) {
    const int b = idx >> 7, h = idx & 127;
    float s = fmaf(ws[WS_TBAR + b * 65 + 64], ws[WS_U + h], ws[WS_C + h]);
    for (int k = 0; k < TED; ++k)
      s = fmaf(ws[WS_TBAR + b * 65 + k], w_key[(FPD + k) * HID + h], s);
    hp[idx] = (s > 0.f) ? s : expm1f(s);             // ELU
  }
  __syncthreads();
  for (int idx = tid; idx < B_ * OUT_; idx += 256) {
    const int b = idx >> 6, o = idx & 63;
    float acc = b_pred[o];
    for (int h = 0; h < HID; ++h) acc = fmaf(hp[b * HID + h], w_pred[h * OUT_ + o], acc);
    y[idx] = acc;
  }
}

// ---------------------------------------------------------------- launch
extern "C" void kernel_launch(void* const* d_in, const int* in_sizes, int n_in,
                              void* d_out, int out_size, void* d_ws, size_t ws_size,
                              hipStream_t stream) {
  const float* key_value  = (const float*)d_in[0];
  const float* key_time   = (const float*)d_in[1];
  const float* query_time = (const float*)d_in[2];
  const float* t2v_w  = (const float*)d_in[3];
  const float* t2v_b  = (const float*)d_in[4];
  const float* w_proj = (const float*)d_in[5];
  const float* b_proj = (const float*)d_in[6];
  const float* w_query= (const float*)d_in[7];
  const float* w_key  = (const float*)d_in[8];
  const float* w_attn = (const float*)d_in[9];
  const float* w_pred = (const float*)d_in[10];
  const float* b_pred = (const float*)d_in[11];

  float* y   = (float*)d_out;                 // [32,64]
  float* att = (float*)d_out + B_ * OUT_;     // [32,16384] (e staged here in-place)
  float* ws  = (float*)d_ws;

  k0_prep<<<1, 128, 0, stream>>>(query_time, t2v_w, t2v_b, w_proj, b_proj, w_query, w_key, ws);
  dim3 g1(N_ / (16 * 8), B_);                 // 128 x 32 blocks, 8 waves each
  k1_logits<<<g1, 256, 0, stream>>>(key_value, key_time, t2v_w, t2v_b, w_key, w_attn, ws, att);
  k2_stats<<<B_, 256, 0, stream>>>(att, ws);
  dim3 g3(8, B_);
  k3_att<<<g3, 256, 0, stream>>>(key_value, key_time, t2v_w, t2v_b, ws, att);
  k4_out<<<1, 256, 0, stream>>>(w_key, w_pred, b_pred, ws, y);
}